// QuantumSuperpositionAttention_85160611545416
// MI455X (gfx1250) — compile-verified
//
#include <hip/hip_runtime.h>
#include <math.h>

// ---------------------------------------------------------------------------
// QuantumSuperpositionAttention — CDNA5 (gfx1250) implementation.
//
// Algebraic optimizations vs reference (all exact for the forward output):
//  * phase MLP + exp(i*phase) are dead code: |beta|^2 == |alpha|^2, and the
//    attention weights are real -> pw1/pb1/pw2/pb2 unused, no sin/cos/silu.
//  * SCALE (DH^-0.5) and any common row factor cancel in probs -> skipped.
//  * only |alpha|^2 needed -> no sqrt.
//
// gfx1250 paths used:
//  * V_WMMA_F32_16X16X4_F32 for all complex GEMMs and attention scores.
//  * GLOBAL_LOAD_ASYNC_TO_LDS_B128 + S_WAIT_ASYNCCNT to stage the shared
//    A-panel per block (removes 8x-redundant per-wave global loads).
//  * GLOBAL_PREFETCH_B8 for the strided B-operand stream.
// ---------------------------------------------------------------------------

typedef float v2f __attribute__((ext_vector_type(2)));
typedef float v8f __attribute__((ext_vector_type(8)));

#define NTOK   1024
#define DMODEL 512
#define NHEAD  8
#define DHEAD  64
#define KSEL   102   // max(1, int(0.1 * 1024))

static __device__ __forceinline__ v8f wmma_f32_16x16x4(v2f a, v2f b, v8f c) {
  // (neg_a, A, neg_b, B, c_mod, C, reuse_a, reuse_b); f32 has no A/B NEG in
  // ISA so we always pass false and negate fragments in VALU when needed.
  return __builtin_amdgcn_wmma_f32_16x16x4_f32(false, a, false, b, (short)0, c,
                                               false, false);
}

// ---------------------------------------------------------------------------
// Cayley: W = (I - A)^-1 (I + A), A = 0.02*(M - M^H).
// One block per weight matrix; thread t owns augmented column t (width 1024).
// I - A is guaranteed well conditioned (A skew-Hermitian), so pivot-free
// Gauss-Jordan matches jnp.linalg.solve numerically well.
// ---------------------------------------------------------------------------
__global__ __launch_bounds__(1024) void cayley_kernel(
    const float* __restrict__ Wq_re, const float* __restrict__ Wq_im,
    const float* __restrict__ Wk_re, const float* __restrict__ Wk_im,
    const float* __restrict__ Wv_re, const float* __restrict__ Wv_im,
    const float* __restrict__ Wo_re, const float* __restrict__ Wo_im,
    float* __restrict__ Gbase, float* __restrict__ Wc) {
  const int mat = blockIdx.x;
  const float* re = (mat == 0) ? Wq_re : (mat == 1) ? Wk_re : (mat == 2) ? Wv_re : Wo_re;
  const float* im = (mat == 0) ? Wq_im : (mat == 1) ? Wk_im : (mat == 2) ? Wv_im : Wo_im;
  float2* G = (float2*)(Gbase + (size_t)mat * 512 * 1024 * 2);
  float* outRe = Wc + (size_t)mat * 2 * 512 * 512;
  float* outIm = outRe + 512 * 512;
  const int tid = threadIdx.x;

  // Build augmented [I - A | I + A].
  for (int idx = tid; idx < 512 * 1024; idx += 1024) {
    int r = idx >> 10, c = idx & 1023;
    int j = (c < 512) ? c : (c - 512);
    float Are = 0.02f * (re[r * 512 + j] - re[j * 512 + r]);
    float Aim = 0.02f * (im[r * 512 + j] + im[j * 512 + r]);
    float eye = (r == j) ? 1.0f : 0.0f;
    float2 g;
    if (c < 512) { g.x = eye - Are; g.y = -Aim; }
    else         { g.x = eye + Are; g.y =  Aim; }
    G[idx] = g;
  }
  __syncthreads();

  __shared__ float2 fac[512];
  __shared__ float2 pvs;
  for (int p = 0; p < 512; ++p) {
    if (tid < 512) fac[tid] = G[tid * 1024 + p];   // pre-scale pivot column
    if (tid == 0)  pvs = G[p * 1024 + p];
    __syncthreads();
    float2 pv = pvs;
    float den = pv.x * pv.x + pv.y * pv.y;
    float2 inv = make_float2(pv.x / den, -pv.y / den);
    float2 g = G[p * 1024 + tid];
    float2 gs = make_float2(g.x * inv.x - g.y * inv.y,
                            g.x * inv.y + g.y * inv.x);
    G[p * 1024 + tid] = gs;
    __syncthreads();
    for (int r = 0; r < 512; ++r) {
      if (r == p) continue;
      float2 f = fac[r];
      float2 t = G[r * 1024 + tid];
      t.x -= f.x * gs.x - f.y * gs.y;
      t.y -= f.x * gs.y + f.y * gs.x;
      G[r * 1024 + tid] = t;
    }
    __syncthreads();
  }

  for (int idx = tid; idx < 512 * 512; idx += 1024) {
    int i = idx >> 9, j = idx & 511;
    float2 g = G[i * 1024 + 512 + j];
    outRe[idx] = g.x;
    outIm[idx] = g.y;
  }
}

// ---------------------------------------------------------------------------
// Complex GEMM C = A * B (planar A/B; C planar or interleaved via ostride).
// Block = 8 waves = one 16(M) x 128(N) macro-tile; K must be 512.
// The block's shared 16x512 A panel (re+im, 64 KB) is staged into LDS with
// GLOBAL_LOAD_ASYNC_TO_LDS_B128 (ASYNCcnt), then read as ds_load_b64 frags.
// A frag: lane(l) -> M = l&15, K-pair = l>>4 ; contiguous v2f LDS load.
// B frag: lane(l) -> N = l&15, K-pair = l>>4 ; two strided global loads.
// C/D   : VGPR r -> M = r + 8*(l>>4), N = l&15.
// ---------------------------------------------------------------------------
__global__ __launch_bounds__(256) void cgemm16(
    const float* __restrict__ Ar, const float* __restrict__ Ai,
    const float* __restrict__ Br, const float* __restrict__ Bi,
    float* __restrict__ Cr, float* __restrict__ Ci,
    int Mm, int Nn, int Kk, int ostride) {
  __shared__ float As[2][16][DMODEL];          // [re/im][row][k] = 64 KB
  const int lane = threadIdx.x & 31;
  const int widx = threadIdx.x >> 5;           // 0..7
  const int blocksN = Nn >> 7;                 // macro tiles along N
  const int tm = (blockIdx.x / blocksN) << 4;
  const int tn = ((blockIdx.x % blocksN) << 7) + (widx << 4);
  const int n15 = lane & 15;
  const int khalf = lane >> 4;

  // --- async-stage A panel: 2 planes * 16 rows * 512 f32 = 4096 b128 ops ---
  {
    const int t = threadIdx.x;
#pragma unroll
    for (int i = 0; i < 16; ++i) {
      int linear = i * 256 + t;                // 0..4095 float4 chunks
      int plane = linear >> 11;                // 0: re, 1: im
      int idx = linear & 2047;
      int row = idx >> 7;                      // 0..15
      int kq = (idx & 127) << 2;               // 0..508, step 4
      const float* src = (plane ? Ai : Ar) + (size_t)(tm + row) * Kk + kq;
      uint32_t loff = (uint32_t)(uintptr_t)&As[plane][row][kq];
      asm volatile("global_load_async_to_lds_b128 %0, %1, off"
                   :: "v"(loff), "v"((uint64_t)(uintptr_t)src)
                   : "memory");
    }
    asm volatile("s_wait_asynccnt 0x0" ::: "memory");
  }
  __syncthreads();

  const float* arow  = &As[0][n15][2 * khalf];
  const float* airow = &As[1][n15][2 * khalf];
  const float* bcol  = Br + tn + n15;
  const float* bicol = Bi + tn + n15;

  v8f accR = {}; v8f accI = {};
  for (int k0 = 0; k0 < Kk; k0 += 4) {
    __builtin_prefetch(bcol + (size_t)(k0 + 8) * Nn, 0, 1);  // global_prefetch_b8
    v2f aR = *(const v2f*)(arow + k0);                       // ds_load_b64
    v2f aI = *(const v2f*)(airow + k0);
    int kr = k0 + 2 * khalf;
    v2f bR = { bcol[(size_t)kr * Nn],  bcol[(size_t)(kr + 1) * Nn] };
    v2f bI = { bicol[(size_t)kr * Nn], bicol[(size_t)(kr + 1) * Nn] };
    v2f aIn = { -aI.x, -aI.y };
    accR = wmma_f32_16x16x4(aR,  bR, accR);   // + Ar*Br
    accR = wmma_f32_16x16x4(aIn, bI, accR);   // - Ai*Bi
    accI = wmma_f32_16x16x4(aR,  bI, accI);   // + Ar*Bi
    accI = wmma_f32_16x16x4(aI,  bR, accI);   // + Ai*Br
  }
#pragma unroll
  for (int r = 0; r < 8; ++r) {
    int row = tm + r + 8 * khalf;
    size_t o = ((size_t)row * Nn + tn + n15) * (size_t)ostride;
    Cr[o] = accR[r];
    Ci[o] = accI[r];
  }
}

// ---------------------------------------------------------------------------
// Attention: per block = (head h, 16 query rows).
// Stage 1: 16 waves build b2[16][1024] = |conj(Q).K|^2 via WMMA (4 s-tiles
//          per wave). B fragments are contiguous loads because K is
//          row-major and the fragment's K-index varies fastest.
// Stage 2: one wave per query row: sum, exact top-102 (ties -> lower index),
//          softmax with reference's 1e-8 semantics, gather+accumulate V.
// ---------------------------------------------------------------------------
__global__ __launch_bounds__(512) void attn_kernel(
    const float* __restrict__ Qr, const float* __restrict__ Qi,
    const float* __restrict__ Kr, const float* __restrict__ Ki,
    const float* __restrict__ Vr, const float* __restrict__ Vi,
    float* __restrict__ AOr, float* __restrict__ AOi,
    float* __restrict__ topv, int* __restrict__ topi) {
  __shared__ float b2[16][NTOK];   // 64 KB
  const int h = blockIdx.x & 7;
  const int q0 = (blockIdx.x >> 3) << 4;
  const int lane = threadIdx.x & 31;
  const int wv = threadIdx.x >> 5;   // 0..15
  const int n15 = lane & 15;
  const int khalf = lane >> 4;
  const size_t hb = (size_t)h * DHEAD;

  // ---- stage 1 ----
  const float* qr = Qr + (size_t)(q0 + n15) * DMODEL + hb + 2 * khalf;
  const float* qi = Qi + (size_t)(q0 + n15) * DMODEL + hb + 2 * khalf;
  for (int st = wv * 4; st < wv * 4 + 4; ++st) {
    const int s0 = st << 4;
    const float* kr = Kr + (size_t)(s0 + n15) * DMODEL + hb + 2 * khalf;
    const float* ki = Ki + (size_t)(s0 + n15) * DMODEL + hb + 2 * khalf;
    v8f accR = {}; v8f accI = {};
    for (int d0 = 0; d0 < DHEAD; d0 += 4) {
      v2f aR = *(const v2f*)(qr + d0);
      v2f aI = *(const v2f*)(qi + d0);
      v2f bR = *(const v2f*)(kr + d0);
      v2f bI = *(const v2f*)(ki + d0);
      v2f aIn = { -aI.x, -aI.y };
      accR = wmma_f32_16x16x4(aR,  bR, accR);   // + Qr.Kr
      accR = wmma_f32_16x16x4(aI,  bI, accR);   // + Qi.Ki   (conj(Q))
      accI = wmma_f32_16x16x4(aR,  bI, accI);   // + Qr.Ki
      accI = wmma_f32_16x16x4(aIn, bR, accI);   // - Qi.Kr
    }
#pragma unroll
    for (int r = 0; r < 8; ++r) {
      int m = r + 8 * khalf;
      float re = accR[r], imv = accI[r];
      b2[m][s0 + n15] = re * re + imv * imv;
    }
  }
  __syncthreads();

  // ---- stage 2 ----
  const int m = wv;
  float* row = b2[m];
  const int gr = blockIdx.x * 16 + m;
  float* tv = topv + (size_t)gr * KSEL;
  int*   ti = topi + (size_t)gr * KSEL;

  float sAll = 0.f;
  for (int t = lane; t < NTOK; t += 32) sAll += row[t];
#pragma unroll
  for (int off = 16; off; off >>= 1) sAll += __shfl_xor(sAll, off, 32);

  float sTop = 0.f, v0 = 0.f;
  for (int j = 0; j < KSEL; ++j) {
    float best = -1.f; int bidx = 0;
    for (int t = lane; t < NTOK; t += 32) {
      float v = row[t];
      if (v > best) { best = v; bidx = t; }
    }
#pragma unroll
    for (int off = 16; off; off >>= 1) {
      float ov = __shfl_xor(best, off, 32);
      int   oi = __shfl_xor(bidx, off, 32);
      if (ov > best || (ov == best && oi < bidx)) { best = ov; bidx = oi; }
    }
    if (lane == 0) { tv[j] = best; ti[j] = bidx; }
    if ((bidx & 31) == lane) row[bidx] = -1.f;   // remove from later rounds
    sTop += best;
    if (j == 0) v0 = best;
  }
  __threadfence_block();

  // softmax(tpn * k): tpn_j = v_j / (sum_top v + 1e-8 * sum_all v)
  const float c = (float)KSEL / (sTop + 1e-8f * sAll);
  float z = 0.f;
  for (int j = lane; j < KSEL; j += 32) z += __expf(c * (tv[j] - v0));
#pragma unroll
  for (int off = 16; off; off >>= 1) z += __shfl_xor(z, off, 32);
  const float invZ = 1.f / z;

  float or0 = 0.f, oi0 = 0.f, or1 = 0.f, oi1 = 0.f;
  for (int j = 0; j < KSEL; ++j) {
    float w = __expf(c * (tv[j] - v0)) * invZ;
    size_t base = (size_t)ti[j] * DMODEL + hb;
    or0 += w * Vr[base + lane];
    oi0 += w * Vi[base + lane];
    or1 += w * Vr[base + lane + 32];
    oi1 += w * Vi[base + lane + 32];
  }
  size_t ob = (size_t)(q0 + m) * DMODEL + hb;
  AOr[ob + lane] = or0;  AOr[ob + lane + 32] = or1;
  AOi[ob + lane] = oi0;  AOi[ob + lane + 32] = oi1;
}

// ---------------------------------------------------------------------------
extern "C" void kernel_launch(void* const* d_in, const int* in_sizes, int n_in,
                              void* d_out, int out_size, void* d_ws,
                              size_t ws_size, hipStream_t stream) {
  const float* x_re  = (const float*)d_in[0];
  const float* x_im  = (const float*)d_in[1];
  const float* Wq_re = (const float*)d_in[2];
  const float* Wq_im = (const float*)d_in[3];
  const float* Wk_re = (const float*)d_in[4];
  const float* Wk_im = (const float*)d_in[5];
  const float* Wv_re = (const float*)d_in[6];
  const float* Wv_im = (const float*)d_in[7];
  const float* Wo_re = (const float*)d_in[8];
  const float* Wo_im = (const float*)d_in[9];
  // d_in[10..13] (pw1,pb1,pw2,pb2) are provably dead in the forward pass.
  (void)in_sizes; (void)n_in; (void)out_size; (void)ws_size;

  // workspace layout (floats); total ~48.7 MB
  float* ws = (float*)d_ws;
  size_t off = 0;
  float* G    = ws + off; off += (size_t)4 * 512 * 1024 * 2;  // augmented solves
  float* Wc   = ws + off; off += (size_t)4 * 2 * 512 * 512;   // cayley results
  float* Qr   = ws + off; off += (size_t)NTOK * DMODEL;
  float* Qi   = ws + off; off += (size_t)NTOK * DMODEL;
  float* Krp  = ws + off; off += (size_t)NTOK * DMODEL;
  float* Kip  = ws + off; off += (size_t)NTOK * DMODEL;
  float* Vrp  = ws + off; off += (size_t)NTOK * DMODEL;
  float* Vip  = ws + off; off += (size_t)NTOK * DMODEL;
  float* AOr  = ws + off; off += (size_t)NTOK * DMODEL;
  float* AOi  = ws + off; off += (size_t)NTOK * DMODEL;
  float* topv = ws + off; off += (size_t)NHEAD * NTOK * KSEL;
  int*   topi = (int*)(ws + off); off += (size_t)NHEAD * NTOK * KSEL;

  const float* WcQr = Wc + 0 * (2 * 512 * 512);
  const float* WcQi = WcQr + 512 * 512;
  const float* WcKr = Wc + 1 * (2 * 512 * 512);
  const float* WcKi = WcKr + 512 * 512;
  const float* WcVr = Wc + 2 * (2 * 512 * 512);
  const float* WcVi = WcVr + 512 * 512;
  const float* WcOr = Wc + 3 * (2 * 512 * 512);
  const float* WcOi = WcOr + 512 * 512;

  cayley_kernel<<<dim3(4), dim3(1024), 0, stream>>>(
      Wq_re, Wq_im, Wk_re, Wk_im, Wv_re, Wv_im, Wo_re, Wo_im, G, Wc);

  // (1024/16) M-tiles * (512/128) N-macro-tiles = 256 blocks of 8 waves
  const dim3 gg(256), gb(256);
  cgemm16<<<gg, gb, 0, stream>>>(x_re, x_im, WcQr, WcQi, Qr, Qi,
                                 NTOK, DMODEL, DMODEL, 1);
  cgemm16<<<gg, gb, 0, stream>>>(x_re, x_im, WcKr, WcKi, Krp, Kip,
                                 NTOK, DMODEL, DMODEL, 1);
  cgemm16<<<gg, gb, 0, stream>>>(x_re, x_im, WcVr, WcVi, Vrp, Vip,
                                 NTOK, DMODEL, DMODEL, 1);

  attn_kernel<<<dim3(NHEAD * (NTOK / 16)), dim3(512), 0, stream>>>(
      Qr, Qi, Krp, Kip, Vrp, Vip, AOr, AOi, topv, topi);

  // final projection, writing interleaved complex64 straight to d_out
  cgemm16<<<gg, gb, 0, stream>>>(AOr, AOi, WcOr, WcOi,
                                 (float*)d_out, (float*)d_out + 1,
                                 NTOK, DMODEL, DMODEL, 2);
}